// GIN_2405181685972
// MI455X (gfx1250) — compile-verified
//
#include <hip/hip_runtime.h>
#include <hip/hip_bf16.h>

// GIN inference on MI455X (gfx1250), wave32, WMMA fp32 path.

typedef float v2f __attribute__((ext_vector_type(2)));
typedef float v8f __attribute__((ext_vector_type(8)));

#define NN 50000
#define EE 800000
#define HH 256
#define LL 4
#define GG 256
#define HP 260          // LDS row stride (floats), padded to avoid bank conflicts
#define BN_EPS 1e-5f

// ---------------------------------------------------------------------------
// abuf = x  (vectorized copy)
__global__ void gin_copy_kernel(const float4* __restrict__ src,
                                float4* __restrict__ dst, int n4) {
    int i = blockIdx.x * blockDim.x + threadIdx.x;
    if (i < n4) dst[i] = src[i];
}

// abuf[dst[e]] += x[src[e]]  (64 threads per edge, float4 gather, f32 atomics)
__global__ void gin_scatter_kernel(const float* __restrict__ X,
                                   const int* __restrict__ esrc,
                                   const int* __restrict__ edst,
                                   float* __restrict__ A) {
    long long tid = (long long)blockIdx.x * blockDim.x + threadIdx.x;
    if (tid >= (long long)EE * 64) return;
    int e = (int)(tid >> 6);
    int c = (int)(tid & 63) << 2;
    int s = esrc[e], d = edst[e];
    float4 v = *(const float4*)(X + (size_t)s * HH + c);
    float* p = A + (size_t)d * HH + c;
    atomicAdd(p + 0, v.x);
    atomicAdd(p + 1, v.y);
    atomicAdd(p + 2, v.z);
    atomicAdd(p + 3, v.w);
}

__global__ void gin_zero_kernel(float* __restrict__ p, int n) {
    int i = blockIdx.x * blockDim.x + threadIdx.x;
    if (i < n) p[i] = 0.0f;
}

// ---------------------------------------------------------------------------
// Fused GIN layer MLP: X' = relu(bn(relu(bn1(A@W1+b1))@W2+b2)); HG += pool(X')
// Block: 256 threads = 8 waves; block owns 64 rows x 256 cols.
// Wave (rg, cg): rg = wave>>2 selects a 32-row group (2 row tiles),
//                cg = wave&3 selects a 64-col group (4 col tiles).
// Per K-step: 2 A-fragment loads + 4 B-fragment loads feed 8 WMMAs.
__global__ __launch_bounds__(256)
void gin_mlp_kernel(const float* __restrict__ A,
                    const float* __restrict__ W1, const float* __restrict__ b1,
                    const float* __restrict__ g1, const float* __restrict__ be1,
                    const float* __restrict__ m1, const float* __restrict__ v1,
                    const float* __restrict__ W2, const float* __restrict__ b2,
                    const float* __restrict__ g2, const float* __restrict__ be2,
                    const float* __restrict__ m2, const float* __restrict__ v2,
                    const int* __restrict__ batch,
                    float* __restrict__ Xout, float* __restrict__ HG, int n) {
    extern __shared__ float z1[];   // [64][HP]
    const int lane = threadIdx.x & 31;
    const int wave = threadIdx.x >> 5;
    const int rg = wave >> 2;       // 32-row group 0..1
    const int cg = wave & 3;        // 64-col group 0..3
    const int mrow = lane & 15;
    const int hi = lane >> 4;       // 0/1: K sub-pair select
    const int row0 = blockIdx.x * 64;

    // ---- GEMM1: z = A @ W1 -------------------------------------------------
    const float* Arow[2];
#pragma unroll
    for (int r = 0; r < 2; ++r) {
        int ar = row0 + rg * 32 + r * 16 + mrow;
        int arc = ar < n ? ar : (n - 1);
        Arow[r] = A + (size_t)arc * HH;
    }

    v8f acc[2][4] = {};
    for (int kk = 0; kk < HH; kk += 4) {
        v2f a[2];
#pragma unroll
        for (int r = 0; r < 2; ++r) {
            a[r].x = Arow[r][kk + 2 * hi];
            a[r].y = Arow[r][kk + 2 * hi + 1];
        }
        v2f b[4];
#pragma unroll
        for (int t = 0; t < 4; ++t) {
            int col = cg * 64 + t * 16 + mrow;
            b[t].x = W1[(size_t)(kk + hi) * HH + col];
            b[t].y = W1[(size_t)(kk + 2 + hi) * HH + col];
        }
#pragma unroll
        for (int r = 0; r < 2; ++r)
#pragma unroll
            for (int t = 0; t < 4; ++t)
                acc[r][t] = __builtin_amdgcn_wmma_f32_16x16x4_f32(
                    false, a[r], false, b[t], (short)0, acc[r][t], false, false);
    }

    // epilogue 1: +b1, BN1, ReLU -> LDS
#pragma unroll
    for (int t = 0; t < 4; ++t) {
        int col = cg * 64 + t * 16 + mrow;
        float sc = g1[col] * rsqrtf(v1[col] + BN_EPS);
        float sh = be1[col] - m1[col] * sc;
        float bb = b1[col];
#pragma unroll
        for (int r = 0; r < 2; ++r)
#pragma unroll
            for (int v = 0; v < 8; ++v) {
                float val = (acc[r][t][v] + bb) * sc + sh;
                val = val > 0.0f ? val : 0.0f;
                int rl = rg * 32 + r * 16 + v + 8 * hi;
                z1[rl * HP + col] = val;
            }
    }
    __syncthreads();

    // ---- GEMM2: z2 = z1 @ W2 ----------------------------------------------
    v8f acc2[2][4] = {};
    const float* Zrow[2];
#pragma unroll
    for (int r = 0; r < 2; ++r)
        Zrow[r] = z1 + (size_t)(rg * 32 + r * 16 + mrow) * HP;

    for (int kk = 0; kk < HH; kk += 4) {
        v2f a[2];
#pragma unroll
        for (int r = 0; r < 2; ++r) {
            a[r].x = Zrow[r][kk + 2 * hi];
            a[r].y = Zrow[r][kk + 2 * hi + 1];
        }
        v2f b[4];
#pragma unroll
        for (int t = 0; t < 4; ++t) {
            int col = cg * 64 + t * 16 + mrow;
            b[t].x = W2[(size_t)(kk + hi) * HH + col];
            b[t].y = W2[(size_t)(kk + 2 + hi) * HH + col];
        }
#pragma unroll
        for (int r = 0; r < 2; ++r)
#pragma unroll
            for (int t = 0; t < 4; ++t)
                acc2[r][t] = __builtin_amdgcn_wmma_f32_16x16x4_f32(
                    false, a[r], false, b[t], (short)0, acc2[r][t], false, false);
    }

    // epilogue 2: +b2, BN, ReLU -> Xout; fused global_add_pool into HG
    int grow[2][8];
    int bg[2][8];
#pragma unroll
    for (int r = 0; r < 2; ++r)
#pragma unroll
        for (int v = 0; v < 8; ++v) {
            grow[r][v] = row0 + rg * 32 + r * 16 + v + 8 * hi;
            bg[r][v] = (grow[r][v] < n) ? batch[grow[r][v]] : 0;
        }
#pragma unroll
    for (int t = 0; t < 4; ++t) {
        int col = cg * 64 + t * 16 + mrow;
        float sc = g2[col] * rsqrtf(v2[col] + BN_EPS);
        float sh = be2[col] - m2[col] * sc;
        float bb = b2[col];
#pragma unroll
        for (int r = 0; r < 2; ++r)
#pragma unroll
            for (int v = 0; v < 8; ++v) {
                float val = (acc2[r][t][v] + bb) * sc + sh;
                val = val > 0.0f ? val : 0.0f;
                if (grow[r][v] < n) {
                    Xout[(size_t)grow[r][v] * HH + col] = val;
                    atomicAdd(&HG[(size_t)bg[r][v] * HH + col], val);
                }
            }
    }
}

// ---------------------------------------------------------------------------
// Head part 1: Hout = relu(HG @ lin1_W + lin1_b), 256x256x256 WMMA GEMM.
// One wave per 16x16 tile; 256 tiles; 32 blocks of 8 waves.
__global__ __launch_bounds__(256)
void gin_head1_kernel(const float* __restrict__ HG, const float* __restrict__ W,
                      const float* __restrict__ bvec, float* __restrict__ Hout) {
    const int wid = blockIdx.x * 8 + (threadIdx.x >> 5);
    const int tm = wid >> 4;      // G/16 = 16 row tiles
    const int tn = wid & 15;      // H/16 = 16 col tiles
    const int lane = threadIdx.x & 31;
    const int mrow = lane & 15;
    const int hi = lane >> 4;

    const float* Arow = HG + (size_t)(tm * 16 + mrow) * HH;
    v8f acc = {};
    for (int kk = 0; kk < HH; kk += 4) {
        v2f a;
        a.x = Arow[kk + 2 * hi];
        a.y = Arow[kk + 2 * hi + 1];
        int col = tn * 16 + mrow;
        v2f b;
        b.x = W[(size_t)(kk + hi) * HH + col];
        b.y = W[(size_t)(kk + 2 + hi) * HH + col];
        acc = __builtin_amdgcn_wmma_f32_16x16x4_f32(
            false, a, false, b, (short)0, acc, false, false);
    }
    int col = tn * 16 + mrow;
    float bb = bvec[col];
#pragma unroll
    for (int v = 0; v < 8; ++v) {
        float val = acc[v] + bb;
        val = val > 0.0f ? val : 0.0f;
        int row = tm * 16 + v + 8 * hi;
        Hout[(size_t)row * HH + col] = val;
    }
}

// Head part 2: out[g] = dot(H[g], cls_W) + cls_b. One wave per graph.
__global__ __launch_bounds__(256)
void gin_head2_kernel(const float* __restrict__ Hin, const float* __restrict__ cw,
                      const float* __restrict__ cb, float* __restrict__ out) {
    int g = blockIdx.x * 8 + (threadIdx.x >> 5);
    int lane = threadIdx.x & 31;
    float s = 0.0f;
    for (int k = lane; k < HH; k += 32) s += Hin[(size_t)g * HH + k] * cw[k];
#pragma unroll
    for (int off = 16; off > 0; off >>= 1) s += __shfl_xor(s, off, 32);
    if (lane == 0) out[g] = s + cb[0];
}

// ---------------------------------------------------------------------------
extern "C" void kernel_launch(void* const* d_in, const int* in_sizes, int n_in,
                              void* d_out, int out_size, void* d_ws, size_t ws_size,
                              hipStream_t stream) {
    const float* x      = (const float*)d_in[0];
    const int*   eidx   = (const int*)d_in[1];     // (2, E) flat
    const int*   batch  = (const int*)d_in[2];
    const float* W1     = (const float*)d_in[3];
    const float* b1     = (const float*)d_in[4];
    const float* g1     = (const float*)d_in[5];
    const float* be1    = (const float*)d_in[6];
    const float* m1     = (const float*)d_in[7];
    const float* v1     = (const float*)d_in[8];
    const float* W2     = (const float*)d_in[9];
    const float* b2     = (const float*)d_in[10];
    const float* g2     = (const float*)d_in[11];
    const float* be2    = (const float*)d_in[12];
    const float* m2     = (const float*)d_in[13];
    const float* v2     = (const float*)d_in[14];
    const float* lin1W  = (const float*)d_in[15];
    const float* lin1b  = (const float*)d_in[16];
    const float* clsW   = (const float*)d_in[17];
    const float* clsb   = (const float*)d_in[18];
    float* out = (float*)d_out;

    const int* esrc = eidx;
    const int* edst = eidx + EE;

    // workspace layout
    float* abuf = (float*)d_ws;                    // N*H
    float* xbuf = abuf + (size_t)NN * HH;          // N*H
    float* hg   = xbuf + (size_t)NN * HH;          // G*H
    float* hh   = hg + (size_t)GG * HH;            // G*H

    // zero graph accumulator
    gin_zero_kernel<<<(GG * HH + 255) / 256, 256, 0, stream>>>(hg, GG * HH);

    const int n4 = NN * HH / 4;
    const int copyBlocks = (n4 + 255) / 256;
    const long long scatThreads = (long long)EE * 64;
    const int scatBlocks = (int)((scatThreads + 255) / 256);
    const int mlpBlocks = (NN + 63) / 64;
    const size_t smem = (size_t)64 * HP * sizeof(float);

    for (int l = 0; l < LL; ++l) {
        const float* xin = (l == 0) ? x : xbuf;
        gin_copy_kernel<<<copyBlocks, 256, 0, stream>>>(
            (const float4*)xin, (float4*)abuf, n4);
        gin_scatter_kernel<<<scatBlocks, 256, 0, stream>>>(xin, esrc, edst, abuf);
        gin_mlp_kernel<<<mlpBlocks, 256, smem, stream>>>(
            abuf,
            W1 + (size_t)l * HH * HH, b1 + (size_t)l * HH,
            g1 + (size_t)l * HH, be1 + (size_t)l * HH,
            m1 + (size_t)l * HH, v1 + (size_t)l * HH,
            W2 + (size_t)l * HH * HH, b2 + (size_t)l * HH,
            g2 + (size_t)l * HH, be2 + (size_t)l * HH,
            m2 + (size_t)l * HH, v2 + (size_t)l * HH,
            batch, xbuf, hg, NN);
    }

    gin_head1_kernel<<<32, 256, 0, stream>>>(hg, lin1W, lin1b, hh);
    gin_head2_kernel<<<32, 256, 0, stream>>>(hh, clsW, clsb, out);
}